// EDModelQP_11141145166269
// MI455X (gfx1250) — compile-verified
//
#include <hip/hip_runtime.h>

// ---------------------------------------------------------------------------
// Batched QP interior-point solver for MI455X (gfx1250, wave32).
// One persistent 8-wave workgroup per batch element runs all 15 IPM steps.
// Dominant FLOPs (G^T diag(d) G, Cholesky updates, block tri-solves, Schur)
// run on v_wmma_f32_16x16x4_f32.  G k-panels are streamed into LDS with the
// Tensor Data Mover (tensor_load_to_lds), double-buffered: panel kp+1 is in
// flight while panel kp feeds the WMMAs; s_wait_tensorcnt(1) retires the
// oldest DMA (TDM ops from one wave complete in order).
// ---------------------------------------------------------------------------

#define NZc    512
#define NEQc   128
#define NINEQc 1024
#define XCc    144          // 128 cols of A^T + 1 rhs col + pad to 9*16
#define NITERc 15
#define EPSc   1e-5f
#define SIGMAc 0.1f
#define REGc   1e-8f

typedef float v2f  __attribute__((ext_vector_type(2)));
typedef float v8f  __attribute__((ext_vector_type(8)));
typedef unsigned int u4v __attribute__((ext_vector_type(4)));
typedef int   i4v  __attribute__((ext_vector_type(4)));
typedef int   i8v  __attribute__((ext_vector_type(8)));

__device__ __forceinline__ v8f wmma4(v2f a, v2f b, v8f c) {
    // (neg_a, A, neg_b, B, c_mod, C, reuse_a, reuse_b)
    return __builtin_amdgcn_wmma_f32_16x16x4_f32(false, a, false, b,
                                                 (short)0, c, false, false);
}

// 1-D TDM copy: nelem fp32 from global gsrc -> LDS byte offset lds_off.
__device__ __forceinline__ void tdm_load_1d(const float* gsrc,
                                            unsigned lds_off, unsigned nelem) {
    unsigned long long ga = (unsigned long long)(const void*)gsrc;
    u4v g0;
    g0[0] = 1u;                                            // count=1, user D#
    g0[1] = lds_off;                                       // lds_addr
    g0[2] = (unsigned)(ga & 0xFFFFFFFFull);                // global_addr[31:0]
    g0[3] = (unsigned)((ga >> 32) & 0x01FFFFFFu) | 0x80000000u; // addr[56:32] | type=2
    i8v g1;
    g1[0] = 0x00020000;                    // data_size=2 (4B), wg_mask=0, no flags
    g1[1] = (int)((nelem & 0xFFFFu) << 16);          // tensor_dim0[15:0]
    g1[2] = (int)(((nelem >> 16) & 0xFFFFu) | (1u << 16)); // dim0[31:16] | tensor_dim1=1
    g1[3] = (int)((nelem & 0xFFFFu) << 16);          // dim1[31:16]=0 | tile_dim0
    g1[4] = 0;                                       // tile_dim1=0, tile_dim2=0
    g1[5] = (int)nelem;                              // tensor_dim0_stride[31:0]
    g1[6] = 0;
    g1[7] = 0;
    i4v gz4; gz4[0] = 0; gz4[1] = 0; gz4[2] = 0; gz4[3] = 0;
    i8v gz8;
    #pragma unroll
    for (int i = 0; i < 8; ++i) gz8[i] = 0;
    // 6-arg form (clang-23 / therock headers): groups 2,3 unused, extra group, cpol
    __builtin_amdgcn_tensor_load_to_lds(g0, g1, gz4, gz4, gz8, 0);
}

__device__ __forceinline__ float wave_sum(float v) {
    #pragma unroll
    for (int o = 16; o > 0; o >>= 1) v += __shfl_down(v, o, 32);
    return v;
}
__device__ __forceinline__ float wave_min(float v) {
    #pragma unroll
    for (int o = 16; o > 0; o >>= 1) v = fminf(v, __shfl_down(v, o, 32));
    return v;
}

// LDS budget (floats)
#define SMEM_FLOATS (16384 + 512 + 128 + 8*1024 + 512 + 128 + 512 + 128 + 1056 + 16)

__global__ void __launch_bounds__(256, 1)
qp_ipm_kernel(const float* __restrict__ Pm,   // (32,512)
              const float* __restrict__ G,    // (1024,512)
              const float* __restrict__ Hv,   // (32,1024)  h
              const float* __restrict__ Am,   // (128,512)
              const float* __restrict__ Bv,   // (32,128)   b
              float* __restrict__ out,        // (32,512)   z
              float* __restrict__ ws) {
    extern __shared__ float smem[];
    float* sPanel  = smem;                 // 16384  (2x8192 TDM ping-pong / chol / S / tiles)
    float* sZ      = sPanel  + 16384;      // 512
    float* sNu     = sZ      + 512;        // 128
    float* sS      = sNu     + 128;        // 1024
    float* sLam    = sS      + 1024;       // 1024
    float* sD      = sLam    + 1024;       // 1024
    float* sT      = sD      + 1024;       // 1024  lam + w
    float* sRineq  = sT      + 1024;       // 1024
    float* sRcomp  = sRineq  + 1024;       // 1024
    float* sDs     = sRcomp  + 1024;       // 1024
    float* sDlam   = sDs     + 1024;       // 1024
    float* sRhsTop = sDlam   + 1024;       // 512
    float* sReq    = sRhsTop + 512;        // 128
    float* sDz     = sReq    + 128;        // 512
    float* sDnu    = sDz     + 512;        // 128
    float* sLd     = sDnu    + 128;        // 1056 (32x33 diag block)
    float* sRed    = sLd     + 1056;       // 16

    const int tid  = threadIdx.x;
    const int wave = tid >> 5;
    const int lane = tid & 31;
    const int lh   = lane >> 4;            // 0/1 : K-half per WMMA f32 layout
    const int lm   = lane & 15;            // M or N within tile
    const int bi   = blockIdx.x;

    const size_t perB = (size_t)NZc*NZc + (size_t)NZc*XCc + (size_t)NEQc*NEQc;
    float* Hb = ws + (size_t)bi * perB;    // 512x512 H / Cholesky factor
    float* Xb = Hb + (size_t)NZc*NZc;      // 512x144 [H^-1 A^T | H^-1 rhs | pad]
    float* Sb = Xb + (size_t)NZc*XCc;      // 128x128 Schur

    const unsigned panelLdsOff = (unsigned)(unsigned long long)(void*)sPanel;

    // ---- init state ----
    for (int i = tid; i < NZc;    i += 256) { sZ[i] = 0.f; sDz[i] = 0.f; }
    for (int i = tid; i < NEQc;   i += 256) { sNu[i] = 0.f; }
    for (int i = tid; i < NINEQc; i += 256) { sS[i] = 1.f; sLam[i] = 1.f; }
    __syncthreads();

    for (int it = 0; it < NITERc; ++it) {
        // ===== P1: rineq = G z + s - h ;  req = A z - b  (wave-per-row) =====
        for (int r = wave * 128; r < wave * 128 + 128; ++r) {
            const float* Gr = G + (size_t)r * NZc;
            float acc = 0.f;
            for (int c = lane; c < NZc; c += 32) acc += Gr[c] * sZ[c];
            acc = wave_sum(acc);
            if (lane == 0) sRineq[r] = acc + sS[r] - Hv[(size_t)bi*NINEQc + r];
        }
        for (int r = wave * 16; r < wave * 16 + 16; ++r) {
            const float* Ar = Am + (size_t)r * NZc;
            float acc = 0.f;
            for (int c = lane; c < NZc; c += 32) acc += Ar[c] * sZ[c];
            acc = wave_sum(acc);
            if (lane == 0) sReq[r] = acc - Bv[(size_t)bi*NEQc + r];
        }
        __syncthreads();

        // ===== P2: mu, d, rcomp, t = lam + (lam*rineq - rcomp)/s =====
        {
            float part = 0.f;
            for (int i = tid; i < NINEQc; i += 256) part += sS[i] * sLam[i];
            part = wave_sum(part);
            if (lane == 0) sRed[wave] = part;
            __syncthreads();
            if (tid == 0) {
                float t = 0.f;
                for (int w = 0; w < 8; ++w) t += sRed[w];
                sRed[8] = t * (1.0f / (float)NINEQc);
            }
            __syncthreads();
            float mu = sRed[8];
            for (int i = tid; i < NINEQc; i += 256) {
                float si = sS[i], li = sLam[i];
                float rc = si * li - SIGMAc * mu;
                sRcomp[i] = rc;
                sD[i] = li / si;
                sT[i] = li + (li * sRineq[i] - rc) / si;
            }
            __syncthreads();
        }

        // ===== P3: rhs_top = -(EPS z + p + nu@A + t@G)  (coalesced columns) =====
        for (int n = tid; n < NZc; n += 256) {
            float acc = 0.f;
            for (int i = 0; i < NINEQc; ++i) acc += sT[i] * G[(size_t)i*NZc + n];
            for (int j = 0; j < NEQc;   ++j) acc += sNu[j] * Am[(size_t)j*NZc + n];
            sRhsTop[n] = -(EPSc * sZ[n] + Pm[(size_t)bi*NZc + n] + acc);
        }
        __syncthreads();

        // ===== P4: H = G^T diag(d) G + EPS I =====
        // WMMA; 16-row G k-panels double-buffered through LDS via TDM.
        for (int nb = 0; nb < NZc; nb += 64) {
            v8f acc[4][4];
            #pragma unroll
            for (int i = 0; i < 4; ++i)
                #pragma unroll
                for (int j = 0; j < 4; ++j)
                    acc[i][j] = (v8f){0.f,0.f,0.f,0.f,0.f,0.f,0.f,0.f};

            __syncthreads();                           // sPanel free from prior use
            if (wave == 0)                             // prologue: panel 0 -> buf A
                tdm_load_1d(G, panelLdsOff, 16 * NZc);

            for (int kp = 0; kp < 64; ++kp) {          // 64 panels x 16 k-rows
                float* cur = sPanel + ((kp & 1) ? 8192 : 0);
                if (wave == 0) {
                    if (kp + 1 < 64) {
                        const unsigned nxtOff =
                            panelLdsOff + (((kp + 1) & 1) ? 32768u : 0u);
                        tdm_load_1d(G + (size_t)(kp + 1) * 16 * NZc, nxtOff, 16 * NZc);
                        __builtin_amdgcn_s_wait_tensorcnt(1);  // oldest (panel kp) done
                    } else {
                        __builtin_amdgcn_s_wait_tensorcnt(0);
                    }
                }
                __syncthreads();                       // panel kp visible to all waves

                const int m0 = wave * 64;
                #pragma unroll
                for (int ks = 0; ks < 4; ++ks) {       // 16 k-rows = 4 WMMA k-steps
                    const int kk = ks * 4 + 2 * lh;
                    const float d0 = sD[kp * 16 + kk];
                    const float d1 = sD[kp * 16 + kk + 1];
                    v2f aF[4], bF[4];
                    #pragma unroll
                    for (int rt = 0; rt < 4; ++rt) {
                        const int col = nb + rt * 16 + lm;
                        aF[rt].x = d0 * cur[kk * NZc + col];
                        aF[rt].y = d1 * cur[(kk + 1) * NZc + col];
                    }
                    #pragma unroll
                    for (int ct = 0; ct < 4; ++ct) {
                        const int col = m0 + ct * 16 + lm;
                        bF[ct].x = cur[kk * NZc + col];
                        bF[ct].y = cur[(kk + 1) * NZc + col];
                    }
                    #pragma unroll
                    for (int rt = 0; rt < 4; ++rt)
                        #pragma unroll
                        for (int ct = 0; ct < 4; ++ct)
                            acc[rt][ct] = wmma4(aF[rt], bF[ct], acc[rt][ct]);
                }
                __syncthreads();                       // all waves done with panel kp
            }
            // store 64x64 block
            #pragma unroll
            for (int rt = 0; rt < 4; ++rt)
                #pragma unroll
                for (int ct = 0; ct < 4; ++ct) {
                    const int n0 = nb + rt * 16;
                    const int m0c = wave * 64 + ct * 16;
                    #pragma unroll
                    for (int r = 0; r < 8; ++r) {
                        const int gi = n0 + r + 8 * lh;
                        const int gj = m0c + lm;
                        float v = acc[rt][ct][r];
                        if (gi == gj) v += EPSc;
                        Hb[(size_t)gi * NZc + gj] = v;
                    }
                }
        }
        __syncthreads();

        // ===== P5: blocked Cholesky of H (lower), 32-wide panels, WMMA trailing =====
        for (int kb = 0; kb < 16; ++kb) {
            const int k0 = kb * 32;
            // diag block -> LDS (lower half; upper mirrored implicitly)
            for (int idx = tid; idx < 32 * 32; idx += 256) {
                int i = idx >> 5, j = idx & 31;
                sLd[i * 33 + j] = (j <= i) ? Hb[(size_t)(k0 + i) * NZc + k0 + j] : 0.f;
            }
            __syncthreads();
            for (int t = 0; t < 32; ++t) {
                if (tid == 0) sLd[t * 33 + t] = sqrtf(fmaxf(sLd[t * 33 + t], 1e-20f));
                __syncthreads();
                const float ltt = sLd[t * 33 + t];
                for (int i = t + 1 + tid; i < 32; i += 256) sLd[i * 33 + t] /= ltt;
                __syncthreads();
                for (int idx = tid; idx < 32 * 32; idx += 256) {
                    int i = idx >> 5, j = idx & 31;
                    if (i > t && j > t && j <= i)
                        sLd[i * 33 + j] -= sLd[i * 33 + t] * sLd[j * 33 + t];
                }
                __syncthreads();
            }
            for (int idx = tid; idx < 32 * 32; idx += 256) {
                int i = idx >> 5, j = idx & 31;
                Hb[(size_t)(k0 + i) * NZc + k0 + j] = (j <= i) ? sLd[i * 33 + j] : 0.f;
            }
            __syncthreads();
            // panel solve: rows below, forward-substitute against diag block
            const int nrows = NZc - k0 - 32;
            for (int r = tid; r < nrows; r += 256) {
                const int gr = k0 + 32 + r;
                float x[32];
                #pragma unroll
                for (int t = 0; t < 32; ++t) {
                    float v = Hb[(size_t)gr * NZc + k0 + t];
                    #pragma unroll
                    for (int u = 0; u < t; ++u) v -= x[u] * sLd[t * 33 + u];
                    x[t] = v / sLd[t * 33 + t];
                }
                #pragma unroll
                for (int t = 0; t < 32; ++t) {
                    Hb[(size_t)gr * NZc + k0 + t] = x[t];
                    sPanel[r * 32 + t] = x[t];
                }
            }
            __syncthreads();
            // trailing SYRK update with WMMA (lower tiles only)
            const int tr0 = k0 + 32;
            const int nt = (NZc - tr0) >> 4;
            int tcount = 0;
            for (int ti = 0; ti < nt; ++ti)
                for (int tj = 0; tj <= ti; ++tj, ++tcount) {
                    if ((tcount & 7) != wave) continue;
                    const int i0l = ti * 16, j0l = tj * 16;
                    v8f acc = {0.f,0.f,0.f,0.f,0.f,0.f,0.f,0.f};
                    #pragma unroll
                    for (int ks = 0; ks < 8; ++ks) {
                        const int kk = ks * 4 + 2 * lh;
                        v2f a, b;
                        a.x = sPanel[(i0l + lm) * 32 + kk];
                        a.y = sPanel[(i0l + lm) * 32 + kk + 1];
                        b.x = sPanel[(j0l + lm) * 32 + kk];
                        b.y = sPanel[(j0l + lm) * 32 + kk + 1];
                        acc = wmma4(a, b, acc);
                    }
                    #pragma unroll
                    for (int r = 0; r < 8; ++r) {
                        const int gi = tr0 + i0l + r + 8 * lh;
                        const int gj = tr0 + j0l + lm;
                        Hb[(size_t)gi * NZc + gj] -= acc[r];
                    }
                }
            __syncthreads();
        }

        // ===== P6: X = L^-T L^-1 [A^T | rhs_top | 0]  (blocked, WMMA GEMMs) =====
        for (int idx = tid; idx < NZc * XCc; idx += 256) {
            const int n = idx / XCc, c = idx - n * XCc;
            float v = 0.f;
            if (c < NEQc) v = Am[(size_t)c * NZc + n];
            else if (c == NEQc) v = sRhsTop[n];
            Xb[idx] = v;
        }
        __syncthreads();
        // forward
        for (int jb = 0; jb < 32; ++jb) {
            const int r0 = jb * 16;
            for (int ct = wave; ct < 9; ct += 8) {
                v8f acc = {0.f,0.f,0.f,0.f,0.f,0.f,0.f,0.f};
                for (int k = 0; k < r0; k += 4) {
                    const int kk = k + 2 * lh;
                    v2f a, b;
                    a.x = Hb[(size_t)(r0 + lm) * NZc + kk];
                    a.y = Hb[(size_t)(r0 + lm) * NZc + kk + 1];
                    b.x = Xb[(size_t)kk * XCc + ct * 16 + lm];
                    b.y = Xb[(size_t)(kk + 1) * XCc + ct * 16 + lm];
                    acc = wmma4(a, b, acc);
                }
                #pragma unroll
                for (int r = 0; r < 8; ++r)
                    sPanel[(r + 8 * lh) * XCc + ct * 16 + lm] = acc[r];
            }
            __syncthreads();
            if (tid < XCc) {
                const int c = tid;
                float x[16];
                #pragma unroll
                for (int t = 0; t < 16; ++t) {
                    float v = Xb[(size_t)(r0 + t) * XCc + c] - sPanel[t * XCc + c];
                    #pragma unroll
                    for (int u = 0; u < t; ++u)
                        v -= Hb[(size_t)(r0 + t) * NZc + r0 + u] * x[u];
                    x[t] = v / Hb[(size_t)(r0 + t) * NZc + r0 + t];
                }
                #pragma unroll
                for (int t = 0; t < 16; ++t) Xb[(size_t)(r0 + t) * XCc + c] = x[t];
            }
            __syncthreads();
        }
        // backward
        for (int jb = 31; jb >= 0; --jb) {
            const int r0 = jb * 16;
            for (int ct = wave; ct < 9; ct += 8) {
                v8f acc = {0.f,0.f,0.f,0.f,0.f,0.f,0.f,0.f};
                for (int k = r0 + 16; k < NZc; k += 4) {
                    const int kk = k + 2 * lh;
                    v2f a, b;
                    a.x = Hb[(size_t)kk * NZc + r0 + lm];        // L[k][r0+M]
                    a.y = Hb[(size_t)(kk + 1) * NZc + r0 + lm];
                    b.x = Xb[(size_t)kk * XCc + ct * 16 + lm];
                    b.y = Xb[(size_t)(kk + 1) * XCc + ct * 16 + lm];
                    acc = wmma4(a, b, acc);
                }
                #pragma unroll
                for (int r = 0; r < 8; ++r)
                    sPanel[(r + 8 * lh) * XCc + ct * 16 + lm] = acc[r];
            }
            __syncthreads();
            if (tid < XCc) {
                const int c = tid;
                float x[16];
                #pragma unroll
                for (int t = 15; t >= 0; --t) {
                    float v = Xb[(size_t)(r0 + t) * XCc + c] - sPanel[t * XCc + c];
                    #pragma unroll
                    for (int u = 15; u > 0; --u)
                        if (u > t) v -= Hb[(size_t)(r0 + u) * NZc + r0 + t] * x[u];
                    x[t] = v / Hb[(size_t)(r0 + t) * NZc + r0 + t];
                }
                #pragma unroll
                for (int t = 0; t < 16; ++t) Xb[(size_t)(r0 + t) * XCc + c] = x[t];
            }
            __syncthreads();
        }

        // ===== P7: Schur S = A X1 + REG I (WMMA) ; rhs_s = A x2 + req =====
        for (int t = wave; t < 64; t += 8) {
            const int i0 = (t >> 3) * 16, j0 = (t & 7) * 16;
            v8f acc = {0.f,0.f,0.f,0.f,0.f,0.f,0.f,0.f};
            for (int k = 0; k < NZc; k += 4) {
                const int kk = k + 2 * lh;
                v2f a, b;
                a.x = Am[(size_t)(i0 + lm) * NZc + kk];
                a.y = Am[(size_t)(i0 + lm) * NZc + kk + 1];
                b.x = Xb[(size_t)kk * XCc + j0 + lm];
                b.y = Xb[(size_t)(kk + 1) * XCc + j0 + lm];
                acc = wmma4(a, b, acc);
            }
            #pragma unroll
            for (int r = 0; r < 8; ++r) {
                const int gi = i0 + r + 8 * lh, gj = j0 + lm;
                float v = acc[r];
                if (gi == gj) v += REGc;
                Sb[(size_t)gi * NEQc + gj] = v;
            }
        }
        for (int r = wave * 16; r < wave * 16 + 16; ++r) {
            const float* Ar = Am + (size_t)r * NZc;
            float acc = 0.f;
            for (int c = lane; c < NZc; c += 32)
                acc += Ar[c] * Xb[(size_t)c * XCc + NEQc];
            acc = wave_sum(acc);
            if (lane == 0) sDnu[r] = acc + sReq[r];
        }
        __syncthreads();

        // ===== P8: Cholesky(S) in LDS + solve for dnu (in sDnu) =====
        for (int idx = tid; idx < NEQc * NEQc; idx += 256) sPanel[idx] = Sb[idx];
        __syncthreads();
        for (int t = 0; t < NEQc; ++t) {
            if (tid == 0) sPanel[t * NEQc + t] = sqrtf(fmaxf(sPanel[t * NEQc + t], 1e-20f));
            __syncthreads();
            const float ltt = sPanel[t * NEQc + t];
            for (int i = t + 1 + tid; i < NEQc; i += 256) sPanel[i * NEQc + t] /= ltt;
            __syncthreads();
            for (int idx = tid; idx < NEQc * NEQc; idx += 256) {
                int i = idx >> 7, j = idx & 127;
                if (i > t && j > t && j <= i)
                    sPanel[idx] -= sPanel[i * NEQc + t] * sPanel[j * NEQc + t];
            }
            __syncthreads();
        }
        for (int t = 0; t < NEQc; ++t) {          // forward
            if (tid == 0) sDnu[t] /= sPanel[t * NEQc + t];
            __syncthreads();
            const float yt = sDnu[t];
            for (int i = t + 1 + tid; i < NEQc; i += 256)
                sDnu[i] -= sPanel[i * NEQc + t] * yt;
            __syncthreads();
        }
        for (int t = NEQc - 1; t >= 0; --t) {     // backward (L^T)
            if (tid == 0) sDnu[t] /= sPanel[t * NEQc + t];
            __syncthreads();
            const float xt = sDnu[t];
            for (int i = tid; i < t; i += 256)
                sDnu[i] -= sPanel[t * NEQc + i] * xt;
            __syncthreads();
        }

        // ===== P9: dz = x2 - X1 dnu =====
        for (int n = tid; n < NZc; n += 256) {
            float acc = Xb[(size_t)n * XCc + NEQc];
            for (int j = 0; j < NEQc; ++j) acc -= Xb[(size_t)n * XCc + j] * sDnu[j];
            sDz[n] = acc;
        }
        __syncthreads();

        // ===== P10: ds, dlam, step length alpha =====
        for (int r = wave * 128; r < wave * 128 + 128; ++r) {
            const float* Gr = G + (size_t)r * NZc;
            float acc = 0.f;
            for (int c = lane; c < NZc; c += 32) acc += Gr[c] * sDz[c];
            acc = wave_sum(acc);
            if (lane == 0) {
                float dsv = -sRineq[r] - acc;
                sDs[r] = dsv;
                sDlam[r] = (-sRcomp[r] - sLam[r] * dsv) / sS[r];
            }
        }
        __syncthreads();
        {
            float amin = 1e10f;
            for (int i = tid; i < NINEQc; i += 256) {
                float dsv = sDs[i], dlv = sDlam[i];
                if (dsv < 0.f) amin = fminf(amin, -sS[i]   / dsv);
                if (dlv < 0.f) amin = fminf(amin, -sLam[i] / dlv);
            }
            amin = wave_min(amin);
            if (lane == 0) sRed[wave] = amin;
            __syncthreads();
            if (tid == 0) {
                float m = 1e10f;
                for (int w = 0; w < 8; ++w) m = fminf(m, sRed[w]);
                sRed[8] = fminf(1.0f, 0.99f * m);
            }
            __syncthreads();
        }
        const float alpha = sRed[8];

        // ===== P11: state update =====
        for (int n = tid; n < NZc;    n += 256) sZ[n]  += alpha * sDz[n];
        for (int j = tid; j < NEQc;   j += 256) sNu[j] += alpha * sDnu[j];
        for (int i = tid; i < NINEQc; i += 256) {
            sS[i]   += alpha * sDs[i];
            sLam[i] += alpha * sDlam[i];
        }
        __syncthreads();
    }

    // ---- write z ----
    for (int n = tid; n < NZc; n += 256) out[(size_t)bi * NZc + n] = sZ[n];
}

extern "C" void kernel_launch(void* const* d_in, const int* in_sizes, int n_in,
                              void* d_out, int out_size, void* d_ws, size_t ws_size,
                              hipStream_t stream) {
    (void)in_sizes; (void)n_in; (void)out_size; (void)ws_size;
    const float* p  = (const float*)d_in[0];   // (32,512)
    const float* G  = (const float*)d_in[1];   // (1024,512)
    const float* h  = (const float*)d_in[2];   // (32,1024)
    const float* A  = (const float*)d_in[3];   // (128,512)
    const float* b  = (const float*)d_in[4];   // (32,128)
    float* out = (float*)d_out;
    float* ws  = (float*)d_ws;                 // needs 32*(512*512+512*144+128*128)*4 ≈ 43 MB

    const int smemBytes = SMEM_FLOATS * (int)sizeof(float);   // ~108 KB (WGP has 320 KB LDS)
    (void)hipFuncSetAttribute((const void*)qp_ipm_kernel,
                              hipFuncAttributeMaxDynamicSharedMemorySize, smemBytes);
    qp_ipm_kernel<<<dim3(32), dim3(256), smemBytes, stream>>>(p, G, h, A, b, out, ws);
}